// GATTreeEncoder_60335700574786
// MI455X (gfx1250) — compile-verified
//
#include <hip/hip_runtime.h>
#include <hip/hip_bf16.h>

// GAT tree encoder for MI455X (gfx1250, wave32).
// Dense GEMMs use V_WMMA_F32_16X16X4_F32 (fp32-exact matrix path).
// Graph scatter/gather uses coalesced wave-per-edge atomics (L2-resident).

typedef __attribute__((ext_vector_type(2))) float v2f;
typedef __attribute__((ext_vector_type(8))) float v8f;

#define NEG_SLOPE 0.2f
#define LN_EPS    1e-5f

// ---------------------------------------------------------------- utilities
__global__ void fill_f32(float* __restrict__ p, long long n, float v) {
  long long i = (long long)blockIdx.x * blockDim.x + threadIdx.x;
  if (i < n) p[i] = v;
}

__device__ __forceinline__ unsigned f2sortable(float f) {
  unsigned u = __float_as_uint(f);
  return (u & 0x80000000u) ? ~u : (u | 0x80000000u);
}
__device__ __forceinline__ float sortable2f(unsigned u) {
  u = (u & 0x80000000u) ? (u & 0x7FFFFFFFu) : ~u;
  return __uint_as_float(u);
}
__device__ __forceinline__ float leaky(float x) {
  return x > 0.0f ? x : NEG_SLOPE * x;
}
__device__ __forceinline__ void edge_sd(const int* __restrict__ ei, int E, int e,
                                        int& s, int& d) {
  if (e < E) { s = ei[e]; d = ei[E + e]; }      // edge_index[0][e], [1][e]
  else       { s = d = e - E; }                 // self-loop
}

// ------------------------------------------------------------ WMMA GEMM
// C[M x 256] = A[M x K] @ B[K x 256]       (MODE 0: plain store)
// MODE 1: C += A@B + bias_a[col] + bias_b[col]   (residual epilogue)
// Block = 256 threads (8 waves); each block -> 16 rows x 256 cols.
// Wave w owns cols [32w, 32w+32) via two 16x16 accumulators.
template <int MODE>
__global__ void gemm_wmma_n256(const float* __restrict__ A,
                               const float* __restrict__ B,
                               float* __restrict__ C,
                               const float* __restrict__ bias_a,
                               const float* __restrict__ bias_b,
                               int M, int K) {
  const int lane  = threadIdx.x & 31;
  const int wave  = threadIdx.x >> 5;
  const int rtile = blockIdx.x;
  const int rlo   = lane & 15;
  const int khalf = (lane >> 4) * 2;           // 0 or 2 (A/B K split per ISA)
  long long arow  = (long long)rtile * 16 + rlo;
  if (arow >= M) arow = M - 1;                 // clamp (no EXEC divergence)
  const int col   = wave * 32 + rlo;

  v8f acc0 = {}; v8f acc1 = {};
  const float* __restrict__ Ar = A + arow * K;
  for (int k = 0; k < K; k += 4) {
    v2f a, b0, b1;
    a.x = Ar[k + khalf];
    a.y = Ar[k + khalf + 1];
    const float* __restrict__ Bk = B + (long long)(k + khalf) * 256;
    b0.x = Bk[col];        b0.y = Bk[256 + col];
    b1.x = Bk[col + 16];   b1.y = Bk[256 + col + 16];
    acc0 = __builtin_amdgcn_wmma_f32_16x16x4_f32(false, a, false, b0,
                                                 (short)0, acc0, false, false);
    acc1 = __builtin_amdgcn_wmma_f32_16x16x4_f32(false, a, false, b1,
                                                 (short)0, acc1, false, false);
  }

  const int rhalf = (lane >> 4) * 8;           // C/D rows r / r+8 per lane half
#pragma unroll
  for (int r = 0; r < 8; ++r) {
    long long orow = (long long)rtile * 16 + rhalf + r;
    if (orow < M) {
      const int c0 = wave * 32 + rlo;
      long long o  = orow * 256 + c0;
      if (MODE == 0) {
        C[o]      = acc0[r];
        C[o + 16] = acc1[r];
      } else {
        C[o]      += acc0[r] + bias_a[c0]      + bias_b[c0];
        C[o + 16] += acc1[r] + bias_a[c0 + 16] + bias_b[c0 + 16];
      }
    }
  }
}

// ------------------------------------------------- per-node attention scores
// as[n,h] = sum_c h[n, h*Cc + c] * att_s[h,c]   (ad likewise). Thread per (n,h).
__global__ void attn_scores(const float* __restrict__ hbuf,
                            const float* __restrict__ att_s,
                            const float* __restrict__ att_d,
                            float* __restrict__ as_out,
                            float* __restrict__ ad_out,
                            int Nn, int Hh, int Cc) {
  long long tid = (long long)blockIdx.x * blockDim.x + threadIdx.x;
  if (tid >= (long long)Nn * Hh) return;
  int n = (int)(tid / Hh), h = (int)(tid % Hh);
  const float* __restrict__ row = hbuf + (long long)n * 256 + h * Cc;
  const float* __restrict__ vs  = att_s + h * Cc;
  const float* __restrict__ vd  = att_d + h * Cc;
  float ss = 0.0f, sd = 0.0f;
  for (int c = 0; c < Cc; ++c) { float v = row[c]; ss += v * vs[c]; sd += v * vd[c]; }
  as_out[(long long)n * Hh + h] = ss;
  ad_out[(long long)n * Hh + h] = sd;
}

// ------------------------------------------------- edge pass 1: segment max
__global__ void edge_max(const int* __restrict__ ei,
                         const float* __restrict__ asrc,
                         const float* __restrict__ adst,
                         unsigned* __restrict__ mbuf,
                         int E, int EA, int Hh) {
  long long tid = (long long)blockIdx.x * blockDim.x + threadIdx.x;
  if (tid >= (long long)EA * Hh) return;
  int e = (int)(tid / Hh), h = (int)(tid % Hh);
  int s, d; edge_sd(ei, E, e, s, d);
  float l = leaky(asrc[(long long)s * Hh + h] + adst[(long long)d * Hh + h]);
  atomicMax(&mbuf[(long long)d * Hh + h], f2sortable(l));
}

// ------------------------------------------------- edge pass 2: sum of exp
__global__ void edge_expsum(const int* __restrict__ ei,
                            const float* __restrict__ asrc,
                            const float* __restrict__ adst,
                            const unsigned* __restrict__ mbuf,
                            float* __restrict__ zbuf,
                            int E, int EA, int Hh) {
  long long tid = (long long)blockIdx.x * blockDim.x + threadIdx.x;
  if (tid >= (long long)EA * Hh) return;
  int e = (int)(tid / Hh), h = (int)(tid % Hh);
  int s, d; edge_sd(ei, E, e, s, d);
  float l = leaky(asrc[(long long)s * Hh + h] + adst[(long long)d * Hh + h]);
  float m = sortable2f(mbuf[(long long)d * Hh + h]);
  atomicAdd(&zbuf[(long long)d * Hh + h], __expf(l - m));
}

// ------------------------------------------------- edge pass 3: aggregate
// Wave per edge; lane covers channel idx = k*32+lane, k=0..7 (256 feats).
__global__ void edge_aggregate(const int* __restrict__ ei,
                               const float* __restrict__ hin,
                               const float* __restrict__ asrc,
                               const float* __restrict__ adst,
                               const unsigned* __restrict__ mbuf,
                               const float* __restrict__ zbuf,
                               float* __restrict__ agg,
                               int E, int EA, int Hh, int Cc) {
  long long gtid = (long long)blockIdx.x * blockDim.x + threadIdx.x;
  int wid  = (int)(gtid >> 5);
  int lane = threadIdx.x & 31;
  if (wid >= EA) return;
  int s, d; edge_sd(ei, E, wid, s, d);
  const float* __restrict__ src_row = hin + (long long)s * 256;
  float* __restrict__ dst_row       = agg + (long long)d * 256;
#pragma unroll
  for (int k = 0; k < 8; ++k) {
    int idx = k * 32 + lane;
    int hh  = idx / Cc;                         // head for this channel
    float l = leaky(asrc[(long long)s * Hh + hh] + adst[(long long)d * Hh + hh]);
    float m = sortable2f(mbuf[(long long)d * Hh + hh]);
    float z = zbuf[(long long)d * Hh + hh];
    float alpha = __expf(l - m) / z;
    atomicAdd(&dst_row[idx], src_row[idx] * alpha);
  }
}

// ------------------------------------------------- LayerNorm(+b1) + ELU
// Wave per node, wave32 shuffle reductions.
__global__ void ln_elu(float* __restrict__ hbuf, const float* __restrict__ b1,
                       const float* __restrict__ gamma,
                       const float* __restrict__ beta, int Nn) {
  long long gtid = (long long)blockIdx.x * blockDim.x + threadIdx.x;
  int n    = (int)(gtid >> 5);
  int lane = threadIdx.x & 31;
  if (n >= Nn) return;
  float* __restrict__ row = hbuf + (long long)n * 256;
  float vals[8];
  float sum = 0.0f;
#pragma unroll
  for (int k = 0; k < 8; ++k) {
    int idx = k * 32 + lane;
    vals[k] = row[idx] + b1[idx];
    sum += vals[k];
  }
#pragma unroll
  for (int off = 16; off > 0; off >>= 1) sum += __shfl_xor(sum, off);
  float mu = sum * (1.0f / 256.0f);
  float var = 0.0f;
#pragma unroll
  for (int k = 0; k < 8; ++k) { float t = vals[k] - mu; var += t * t; }
#pragma unroll
  for (int off = 16; off > 0; off >>= 1) var += __shfl_xor(var, off);
  var *= (1.0f / 256.0f);
  float rs = rsqrtf(var + LN_EPS);
#pragma unroll
  for (int k = 0; k < 8; ++k) {
    int idx = k * 32 + lane;
    float t = (vals[k] - mu) * rs * gamma[idx] + beta[idx];
    row[idx] = t > 0.0f ? t : (__expf(t) - 1.0f);   // ELU
  }
}

// ------------------------------------------------- global mean pool
__global__ void pool_accum(const float* __restrict__ o,
                           const int* __restrict__ batch,
                           float* __restrict__ outsum,
                           float* __restrict__ cnt, int Nn) {
  long long gtid = (long long)blockIdx.x * blockDim.x + threadIdx.x;
  int n    = (int)(gtid >> 5);
  int lane = threadIdx.x & 31;
  if (n >= Nn) return;
  int g = batch[n];
  const float* __restrict__ row = o + (long long)n * 256;
  float* __restrict__ acc       = outsum + (long long)g * 256;
#pragma unroll
  for (int k = 0; k < 8; ++k) {
    int idx = k * 32 + lane;
    atomicAdd(&acc[idx], row[idx]);
  }
  if (lane == 0) atomicAdd(&cnt[g], 1.0f);
}

__global__ void pool_div(float* __restrict__ out, const float* __restrict__ cnt,
                         int total) {
  int i = blockIdx.x * blockDim.x + threadIdx.x;
  if (i >= total) return;
  out[i] /= fmaxf(cnt[i >> 8], 1.0f);
}

// ---------------------------------------------------------------- driver
static inline int cdiv(long long a, long long b) { return (int)((a + b - 1) / b); }

extern "C" void kernel_launch(void* const* d_in, const int* in_sizes, int n_in,
                              void* d_out, int out_size, void* d_ws, size_t ws_size,
                              hipStream_t stream) {
  const float* x        = (const float*)d_in[0];
  const int*   ei       = (const int*)  d_in[1];
  const int*   batch    = (const int*)  d_in[2];
  const float* W1       = (const float*)d_in[3];
  const float* att_src1 = (const float*)d_in[4];
  const float* att_dst1 = (const float*)d_in[5];
  const float* b1       = (const float*)d_in[6];
  const float* gamma    = (const float*)d_in[7];
  const float* beta     = (const float*)d_in[8];
  const float* W2       = (const float*)d_in[9];
  const float* att_src2 = (const float*)d_in[10];
  const float* att_dst2 = (const float*)d_in[11];
  const float* b2       = (const float*)d_in[12];
  const float* Wres     = (const float*)d_in[13];
  const float* bres     = (const float*)d_in[14];
  float* out = (float*)d_out;

  const int Nn = in_sizes[2];          // 50000 (batch vector length)
  const int Ee = in_sizes[1] / 2;      // 800000
  const int Gg = out_size / 256;       // 512
  const int EA = Ee + Nn;              // edges + self-loops
  const int IN_DIM = in_sizes[0] / Nn; // 128

  // Workspace layout (floats). ~161 MB total.
  const long long NB = (long long)Nn * 256;
  float* ws   = (float*)d_ws;
  float* buf0 = ws;                    // h1, later h2
  float* buf1 = ws + NB;               // agg1 -> h (post LN+ELU)
  float* buf2 = ws + 2 * NB;           // agg2 -> o
  float* scal = ws + 3 * NB;           // zeroed scalar region
  unsigned* m1 = (unsigned*)scal;                    // N*8
  float*    z1 = scal + (long long)Nn * 8;           // N*8
  unsigned* m2 = (unsigned*)(scal + (long long)Nn * 16); // N
  float*    z2 = scal + (long long)Nn * 17;          // N
  float*    cnt = scal + (long long)Nn * 18;         // G
  float*    as1 = cnt + Gg;                          // N*8
  float*    ad1 = as1 + (long long)Nn * 8;           // N*8
  float*    as2 = ad1 + (long long)Nn * 8;           // N
  float*    ad2 = as2 + Nn;                          // N

  const int B = 256;
  const long long zlen1 = 2 * NB;                    // buf1 + buf2
  const long long zlen2 = (long long)Nn * 18 + Gg;   // m/z/cnt

  // 1) zero accumulators
  fill_f32<<<cdiv(zlen1, B), B, 0, stream>>>(buf1, zlen1, 0.0f);
  fill_f32<<<cdiv(zlen2, B), B, 0, stream>>>(scal, zlen2, 0.0f);
  fill_f32<<<cdiv((long long)Gg * 256, B), B, 0, stream>>>(out, (long long)Gg * 256, 0.0f);

  const int gtiles = cdiv(Nn, 16);

  // 2) h1 = x @ W1
  gemm_wmma_n256<0><<<gtiles, B, 0, stream>>>(x, W1, buf0, nullptr, nullptr, Nn, IN_DIM);
  // 3) attention scores layer 1
  attn_scores<<<cdiv((long long)Nn * 8, B), B, 0, stream>>>(buf0, att_src1, att_dst1,
                                                            as1, ad1, Nn, 8, 32);
  // 4) softmax + aggregate layer 1
  edge_max   <<<cdiv((long long)EA * 8, B), B, 0, stream>>>(ei, as1, ad1, m1, Ee, EA, 8);
  edge_expsum<<<cdiv((long long)EA * 8, B), B, 0, stream>>>(ei, as1, ad1, m1, z1, Ee, EA, 8);
  edge_aggregate<<<cdiv((long long)EA * 32, B), B, 0, stream>>>(ei, buf0, as1, ad1, m1, z1,
                                                                buf1, Ee, EA, 8, 32);
  // 5) LayerNorm(+b1) + ELU in place -> h
  ln_elu<<<cdiv((long long)Nn * 32, B), B, 0, stream>>>(buf1, b1, gamma, beta, Nn);

  // 6) h2 = h @ W2
  gemm_wmma_n256<0><<<gtiles, B, 0, stream>>>(buf1, W2, buf0, nullptr, nullptr, Nn, 256);
  attn_scores<<<cdiv((long long)Nn, B), B, 0, stream>>>(buf0, att_src2, att_dst2,
                                                        as2, ad2, Nn, 1, 256);
  edge_max   <<<cdiv((long long)EA, B), B, 0, stream>>>(ei, as2, ad2, m2, Ee, EA, 1);
  edge_expsum<<<cdiv((long long)EA, B), B, 0, stream>>>(ei, as2, ad2, m2, z2, Ee, EA, 1);
  edge_aggregate<<<cdiv((long long)EA * 32, B), B, 0, stream>>>(ei, buf0, as2, ad2, m2, z2,
                                                                buf2, Ee, EA, 1, 256);
  // 7) o = agg2 + b2 + h @ Wres + bres  (fused epilogue)
  gemm_wmma_n256<1><<<gtiles, B, 0, stream>>>(buf1, Wres, buf2, b2, bres, Nn, 256);

  // 8) global mean pool
  pool_accum<<<cdiv((long long)Nn * 32, B), B, 0, stream>>>(buf2, batch, out, cnt, Nn);
  pool_div  <<<cdiv((long long)Gg * 256, B), B, 0, stream>>>(out, cnt, Gg * 256);
}